// PrototypicalClassifier_70617852281322
// MI455X (gfx1250) — compile-verified
//
#include <hip/hip_runtime.h>
#include <hip/hip_bf16.h>

// MI455X / gfx1250, wave32. Memory-bound problem (~14MB traffic, 0.54 GFLOP)
// -> keep f32 end-to-end, V_WMMA_F32_16X16X4_F32 for the 8192x256x128 GEMM,
//    TDM (tensor_load_to_lds + TENSORcnt) for prototype staging into LDS.

typedef __attribute__((ext_vector_type(2))) float v2f;
typedef __attribute__((ext_vector_type(8))) float v8f;
typedef __attribute__((ext_vector_type(4))) unsigned int v4u;
typedef __attribute__((ext_vector_type(8))) int v8i;
typedef __attribute__((ext_vector_type(4))) int v4i;

#define N_CLASSES 256
#define DIM       128   // K
#define M_ROWS    8192
#define NCHUNK    64    // prototypes staged per LDS chunk (32 KB)

// ---------------------------------------------------------------------------
// Kernel 1: per-class prototypes + squared norms. Deterministic (no atomics).
// block = class id (256 blocks), thread = feature dim (128 threads).
// ---------------------------------------------------------------------------
__global__ void __launch_bounds__(DIM)
proto_kernel(const float* __restrict__ support,   // [4096][128]
             const int*   __restrict__ labels,    // [4096]
             int n_support,
             float* __restrict__ protos,          // [256][128]
             float* __restrict__ pnorm)           // [256]
{
    const int c = blockIdx.x;
    const int t = threadIdx.x;

    float sum = 0.f;
    int count = 0;
    for (int i = 0; i < n_support; ++i) {
        // labels[i] is wave-uniform -> scalar load; branch is s_cbranch.
        if (labels[i] == c) {
            sum += support[(size_t)i * DIM + t];
            ++count;
        }
    }
    const float p = sum / fmaxf((float)count, 1.f);   // matches reference sums/max(counts,1)
    protos[(size_t)c * DIM + t] = p;

    __shared__ float red[DIM];
    red[t] = p * p;
    __syncthreads();
    #pragma unroll
    for (int s = DIM / 2; s > 0; s >>= 1) {
        if (t < s) red[t] += red[t + s];
        __syncthreads();
    }
    if (t == 0) pnorm[c] = red[0];
}

// ---------------------------------------------------------------------------
// Kernel 2: out[m][n] = 2*dot(x_m, p_n) - |x_m|^2 - |p_n|^2  via f32 WMMA.
// Block = 128 threads = 4 waves; each wave owns a 16-row M tile (block: 64 rows).
// Prototypes staged into LDS by the Tensor Data Mover (one DMA per chunk).
// ---------------------------------------------------------------------------
__global__ void __launch_bounds__(128)
dist_kernel(const float* __restrict__ x,       // [8192][128]
            const float* __restrict__ protos,  // [256][128]
            const float* __restrict__ pnorm,   // [256]
            float* __restrict__ out)           // [8192][256]
{
    __shared__ float lds_p[NCHUNK * DIM];   // 32 KB prototype chunk (TDM target)
    __shared__ float lds_pn[NCHUNK];        // chunk of |p|^2
    __shared__ float lds_xn[4 * 16];        // per-wave row norms |x_m|^2

    const int tid  = threadIdx.x;
    const int wave = tid >> 5;
    const int lane = tid & 31;
    const int hi   = lane >> 4;   // half-wave select
    const int lrow = lane & 15;

    const int m0 = blockIdx.x * 64 + wave * 16;   // wave's M-tile base row

    // --- A fragments, ISA 32-bit 16x4 layout ---------------------------------
    // lane<16 : VGPR0=K(4kk+0), VGPR1=K(4kk+1);  lane>=16: K(4kk+2), K(4kk+3)
    // Both half-waves carry row M = lane&15.
    v2f a[DIM / 4];
    const float* xrow = x + (size_t)(m0 + lrow) * DIM + hi * 2;
    float xn_part = 0.f;
    #pragma unroll
    for (int kk = 0; kk < DIM / 4; ++kk) {
        v2f t = *(const v2f*)(xrow + kk * 4);
        a[kk] = t;
        xn_part = fmaf(t.x, t.x, fmaf(t.y, t.y, xn_part));
    }
    // lanes L and L+16 hold complementary K-halves of row L -> combine.
    float xn = xn_part + __shfl_xor(xn_part, 16, 32);
    if (hi == 0) lds_xn[wave * 16 + lrow] = xn;

    for (int nc = 0; nc < N_CLASSES / NCHUNK; ++nc) {
        __syncthreads();
        // ---- TDM: DMA one 64x128 f32 tile of prototypes into LDS ------------
        // D# per CDNA5 ISA ch.8: group0 = {flags, lds_addr, global_addr, type},
        // group1 = {mask/data_size, dims, tile dims, strides}. 2-D tile, so
        // groups 2/3 (and the trailing group) are unused (zero). Issued by
        // wave 0 only; EXEC is ignored by tensor ops; completion tracked with
        // TENSORcnt (s_wait_tensorcnt 0), then workgroup barrier publishes LDS.
        if (wave == 0) {
            const unsigned lds_base = (unsigned)(size_t)(void*)lds_p;       // LDS offset = addr[31:0]
            const unsigned long long ga =
                (unsigned long long)(size_t)(protos + (size_t)nc * NCHUNK * DIM);
            v4u g0;
            g0[0] = 1u;                                        // count=1, user descriptor
            g0[1] = lds_base;                                  // lds_addr (bytes)
            g0[2] = (unsigned)(ga & 0xFFFFFFFFu);              // global_addr[31:0]
            g0[3] = (unsigned)((ga >> 32) & 0x01FFFFFFu)       // global_addr[56:32]
                    | (2u << 30);                              // type=2 ("image")
            v8i g1;
            g1[0] = (int)(2u << 16);            // workgroup_mask=0, data_size=2 (4B)
            g1[1] = (int)(DIM << 16);           // tensor_dim0[15:0]=128  (bits 63:48)
            g1[2] = (int)(N_CLASSES << 16);     // tensor_dim0 hi=0 | tensor_dim1[15:0]=256
            g1[3] = (int)(DIM << 16);           // tensor_dim1 hi=0 | tile_dim0=128
            g1[4] = (int)NCHUNK;                // tile_dim1=64, tile_dim2=0 (2-D)
            g1[5] = (int)DIM;                   // tensor_dim0_stride[31:0]=128 elements
            g1[6] = 0;                          // stride hi | tensor_dim1_stride lo (unused)
            g1[7] = 0;                          // tensor_dim1_stride hi (unused)
            const v4i gz4 = {0, 0, 0, 0};
            const v8i gz8 = {0, 0, 0, 0, 0, 0, 0, 0};
            __builtin_amdgcn_tensor_load_to_lds(g0, g1, gz4, gz4, gz8, 0);
            __builtin_amdgcn_s_wait_tensorcnt(0);
        }
        if (tid < NCHUNK) lds_pn[tid] = pnorm[nc * NCHUNK + tid];
        // gfx1250 global_prefetch_b8: warm L2 with the next chunk.
        if (nc + 1 < N_CLASSES / NCHUNK)
            __builtin_prefetch(protos + (size_t)(nc + 1) * NCHUNK * DIM + tid * 16, 0, 1);
        __syncthreads();

        #pragma unroll
        for (int j = 0; j < NCHUNK / 16; ++j) {       // 16-wide N tiles
            v8f acc = {};
            // B fragment (4x16, B[k][n] = P[n][k]): lane n=lrow holds column n;
            // lane<16 -> K(4kk+0..1), lane>=16 -> K(4kk+2..3).
            const float* bp = lds_p + (size_t)(j * 16 + lrow) * DIM + hi * 2;
            #pragma unroll
            for (int kk = 0; kk < DIM / 4; ++kk) {
                v2f b = *(const v2f*)(bp + kk * 4);
                acc = __builtin_amdgcn_wmma_f32_16x16x4_f32(
                        /*neg_a=*/false, a[kk],
                        /*neg_b=*/false, b,
                        /*c_mod=*/(short)0, acc,
                        /*reuse_a=*/false, /*reuse_b=*/false);
            }
            // Epilogue: C/D layout -> VGPR r: lanes 0-15 hold M=r, lanes 16-31 M=8+r;
            // N = tile_base + (lane&15).
            const int n = nc * NCHUNK + j * 16 + lrow;
            const float pn = lds_pn[j * 16 + lrow];
            float* orow = out + (size_t)(m0 + hi * 8) * N_CLASSES + n;
            #pragma unroll
            for (int r = 0; r < 8; ++r) {
                const float xnr = lds_xn[wave * 16 + hi * 8 + r];
                orow[(size_t)r * N_CLASSES] = fmaf(2.f, acc[r], -(xnr + pn));
            }
        }
    }
}

// ---------------------------------------------------------------------------
extern "C" void kernel_launch(void* const* d_in, const int* in_sizes, int n_in,
                              void* d_out, int out_size, void* d_ws, size_t ws_size,
                              hipStream_t stream) {
    const float* x       = (const float*)d_in[0];   // [8192,128] f32
    const float* support = (const float*)d_in[1];   // [4096,128] f32
    const int*   labels  = (const int*)d_in[2];     // [4096] i32

    const int n_support = in_sizes[1] / DIM;

    float* protos = (float*)d_ws;                   // 256*128 f32 = 128 KB
    float* pnorm  = protos + N_CLASSES * DIM;       // 256 f32

    proto_kernel<<<N_CLASSES, DIM, 0, stream>>>(support, labels, n_support, protos, pnorm);
    dist_kernel<<<M_ROWS / 64, 128, 0, stream>>>(x, protos, pnorm, (float*)d_out);
}